// CGMPBlock_28741921145489
// MI455X (gfx1250) — compile-verified
//
#include <hip/hip_runtime.h>

typedef __attribute__((ext_vector_type(2))) float v2f;
typedef __attribute__((ext_vector_type(8))) float v8f;

// ---------------- CG table metadata ----------------
// 15 (l1,l2,lo) tables, dense layout [2lo+1][2l1+1][2l2+1], concatenated (615 f32).
__constant__ int TBL_L1[15]  = {0,1,2, 0,1,1,1,2,2, 0,1,1,2,2,2};
__constant__ int TBL_L2[15]  = {0,1,2, 1,0,1,2,1,2, 2,1,2,0,1,2};
__constant__ int TBL_LO[15]  = {0,0,0, 1,1,1,1,1,1, 2,2,2,2,2,2};
__constant__ int TBL_OFF[16] = {0,1,10,35,44,53,80,125,170,245,270,315,390,415,490,615};

// ---------------- CG coefficient init (device, per launch: deterministic) ----------------
__device__ double dfact(int n) { double r = 1.0; for (int i = 2; i <= n; ++i) r *= (double)i; return r; }

__device__ float cg_coeff(int j1, int m1, int j2, int m2, int j, int m) {
  if (m != m1 + m2 || j < abs(j1 - j2) || j > j1 + j2) return 0.0f;
  double pre = sqrt((double)(2 * j + 1) * dfact(j + j1 - j2) * dfact(j - j1 + j2) *
                    dfact(j1 + j2 - j) / dfact(j1 + j2 + j + 1));
  pre *= sqrt(dfact(j + m) * dfact(j - m) * dfact(j1 - m1) * dfact(j1 + m1) *
              dfact(j2 - m2) * dfact(j2 + m2));
  double s = 0.0;
  for (int k = 0; k <= j1 + j2 - j; ++k) {
    int d0 = k, d1 = j1 + j2 - j - k, d2 = j1 - m1 - k;
    int d3 = j2 + m2 - k, d4 = j - j2 + m1 + k, d5 = j - j1 - m2 + k;
    if (d0 < 0 || d1 < 0 || d2 < 0 || d3 < 0 || d4 < 0 || d5 < 0) continue;
    double denom = dfact(d0) * dfact(d1) * dfact(d2) * dfact(d3) * dfact(d4) * dfact(d5);
    s += ((k & 1) ? -1.0 : 1.0) / denom;
  }
  return (float)(pre * s);
}

__global__ void cg_init(float* __restrict__ CG) {
  int t = blockIdx.x * blockDim.x + threadIdx.x;
  if (t >= 615) return;
  int ti = 0;
  while (t >= TBL_OFF[ti + 1]) ++ti;
  int r = t - TBL_OFF[ti];
  int l1 = TBL_L1[ti], l2 = TBL_L2[ti], lo = TBL_LO[ti];
  int M1 = 2 * l1 + 1, M2 = 2 * l2 + 1;
  int m   = r / (M1 * M2);
  int rem = r % (M1 * M2);
  int m1 = rem / M2, m2 = rem % M2;
  CG[t] = cg_coeff(l1, m1 - l1, l2, m2 - l2, lo, m - lo);
}

// ---------------- weight relayout: Wcat[lo] -> WMMA 4x16 B-fragment order ----------------
// Global k-step s = 0..419 (lo0: 0..83, lo1: 84..251, lo2: 252..419) + 2 zero pad steps.
// B frag layout (4x16 f32): lane l, vgpr v holds B[K = v + 2*(l>>4)][N = l&15].
__global__ void wb_build(float* __restrict__ WB,
                         const float* __restrict__ Wyy0, const float* __restrict__ Wyy1, const float* __restrict__ Wyy2,
                         const float* __restrict__ Wyx0, const float* __restrict__ Wyx1, const float* __restrict__ Wyx2,
                         const float* __restrict__ Wxx0, const float* __restrict__ Wxx1, const float* __restrict__ Wxx2) {
  int gid = blockIdx.x * blockDim.x + threadIdx.x;
  if (gid >= 422 * 32) return;          // last 64 threads write the 2 pipeline pad fragments
  int s = gid >> 5, lane = gid & 31;
  if (s >= 420) {                       // zero pad so B-prefetch may overrun by two k-steps
    WB[(size_t)(s * 32 + lane) * 2 + 0] = 0.0f;
    WB[(size_t)(s * 32 + lane) * 2 + 1] = 0.0f;
    return;
  }
  int lo, ls;
  if (s < 84)      { lo = 0; ls = s; }
  else if (s < 252){ lo = 1; ls = s - 84; }
  else             { lo = 2; ls = s - 252; }
  const int Kyy[3] = {192, 384, 384}, Kyx[3] = {96, 192, 192};
  const float* Wyy[3] = {Wyy0, Wyy1, Wyy2};
  const float* Wyx[3] = {Wyx0, Wyx1, Wyx2};
  const float* Wxx[3] = {Wxx0, Wxx1, Wxx2};
  int col = lane & 15;
  int kb  = ls * 4 + ((lane >> 4) << 1);
  for (int v = 0; v < 2; ++v) {
    int k = kb + v;
    float val = 0.0f;
    if (col < 8) {
      if (k < Kyy[lo])                 val = Wyy[lo][k * 8 + col];
      else if (k < Kyy[lo] + Kyx[lo])  val = Wyx[lo][(k - Kyy[lo]) * 8 + col];
      else                             val = Wxx[lo][(k - Kyy[lo] - Kyx[lo]) * 8 + col];
    }
    WB[(size_t)(s * 32 + lane) * 2 + v] = val;
  }
}

// ---------------- edge scatter (mp_layer): y[dst] += ev ⊗ x[src] ----------------
// Y layout per node (72 f32): l0 @0 (8ch*1), l1 @8 (8ch*3), l2 @32 (8ch*5); ch = ech*4 + c.
__global__ void scatter_mp(const float* __restrict__ x0, const float* __restrict__ x1,
                           const float* __restrict__ x2, const float* __restrict__ ev,
                           const int* __restrict__ eidx, float* __restrict__ Y,
                           int N, int E) {
  int e = blockIdx.x * blockDim.x + threadIdx.x;
  if (e >= E) return;
  int s = eidx[e];
  int d = eidx[E + e];
  float e0 = ev[2 * e], e1 = ev[2 * e + 1];
  float* yd = Y + (size_t)d * 72;
#pragma unroll
  for (int c = 0; c < 4; ++c) {
    float v = x0[(size_t)s * 4 + c];
    atomicAdd(yd + c,     e0 * v);
    atomicAdd(yd + 4 + c, e1 * v);
  }
#pragma unroll
  for (int c = 0; c < 4; ++c)
#pragma unroll
    for (int m = 0; m < 3; ++m) {
      float v = x1[(size_t)s * 12 + c * 3 + m];
      atomicAdd(yd + 8 + c * 3 + m,      e0 * v);
      atomicAdd(yd + 8 + 12 + c * 3 + m, e1 * v);
    }
#pragma unroll
  for (int c = 0; c < 4; ++c)
#pragma unroll
    for (int m = 0; m < 5; ++m) {
      float v = x2[(size_t)s * 20 + c * 5 + m];
      atomicAdd(yd + 32 + c * 5 + m,      e0 * v);
      atomicAdd(yd + 32 + 20 + c * 5 + m, e1 * v);
    }
}

// ---------------- fused CG-product + SO(3)-linear via v_wmma_f32_16x16x4_f32 ----------------
#define WPB 4                                   // waves per block
#define SLAB_F (16*72 + 16*36 + 5*16*64)        // ytile + xtile + z buffer = 6848 f32 per wave

// Wave-local LDS fence: CDNA5 LDS ops are in-order per wave; we only need the
// DScnt drained + a compiler barrier before cross-lane LDS reuse within the wave.
__device__ __forceinline__ void lds_fence_wave() {
  asm volatile("s_wait_dscnt 0x0" ::: "memory");
}

// One CG chunk, fully specialized: compute z[m][node][c*CB+d] in the wave's LDS
// slab, then sweep it with v_wmma_f32_16x16x4_f32 against the pre-relaid weight
// fragments. All shapes, CG offsets and WB k-step bases are compile-time.
// The B-fragment stream is explicitly double-buffered two k-steps ahead so the
// WMMAs never wait on same-iteration global loads.
template<int LO, int L1, int L2, int AK, int BK, int CGO, int KSB>
__device__ __forceinline__ void process_chunk(const float* __restrict__ cg_s,
                                              const float* __restrict__ yt,
                                              const float* __restrict__ xt,
                                              float* __restrict__ zt,
                                              const float* __restrict__ WB,
                                              int lane, v8f* __restrict__ acc) {
  constexpr int MLO = 2 * LO + 1;
  constexpr int M1  = 2 * L1 + 1;
  constexpr int M2  = 2 * L2 + 1;
  constexpr int CB  = BK ? 4 : 8;
  constexpr int CA  = AK ? 4 : 8;
  constexpr int CC  = CA * CB;
  constexpr int NKS = CC / 4;                   // always even (4, 8 or 16)
  constexpr int AOFF = AK ? (L1 == 0 ? 0 : L1 == 1 ? 4 : 16) : (L1 == 0 ? 0 : L1 == 1 ? 8 : 32);
  constexpr int BOFF = BK ? (L2 == 0 ? 0 : L2 == 1 ? 4 : 16) : (L2 == 0 ? 0 : L2 == 1 ? 8 : 32);
  const int half = lane >> 4;
  const int nl   = lane & 15;

  // CG coefficients for this chunk -> registers (selection rule m = m1 + m2).
  float g[M1 * M2];
#pragma unroll
  for (int m1 = 0; m1 < M1; ++m1)
#pragma unroll
    for (int m2 = 0; m2 < M2; ++m2) {
      const int mo = (m1 - L1) + (m2 - L2) + LO;
      g[m1 * M2 + m2] = (mo >= 0 && mo < MLO)
                          ? cg_s[CGO + (mo * M1 + m1) * M2 + m2] : 0.0f;
    }

  const float* ab = (AK ? xt + nl * 36 : yt + nl * 72) + AOFF;
  const float* bb = (BK ? xt + nl * 36 : yt + nl * 72) + BOFF;

  // ---- sparse CG bilinear (each lane-half covers half the (c,d) space) ----
#pragma unroll 1
  for (int cdi = 0; cdi < CC / 2; ++cdi) {
    const int cd = half * (CC / 2) + cdi;
    const int ci = cd / CB;
    const int d  = cd % CB;
    float av[M1], bv[M2];
#pragma unroll
    for (int m1 = 0; m1 < M1; ++m1) av[m1] = ab[ci * M1 + m1];
#pragma unroll
    for (int m2 = 0; m2 < M2; ++m2) bv[m2] = bb[d * M2 + m2];
    float zz[MLO];
#pragma unroll
    for (int m = 0; m < MLO; ++m) zz[m] = 0.0f;
#pragma unroll
    for (int m1 = 0; m1 < M1; ++m1)
#pragma unroll
      for (int m2 = 0; m2 < M2; ++m2) {
        const int mo = (m1 - L1) + (m2 - L2) + LO;
        if (mo >= 0 && mo < MLO) zz[mo] += g[m1 * M2 + m2] * av[m1] * bv[m2];
      }
#pragma unroll
    for (int m = 0; m < MLO; ++m) zt[(m * 16 + nl) * CC + cd] = zz[m];
  }
  lds_fence_wave();

  // ---- WMMA sweep: two k-steps per iteration, B prefetched one iteration ahead ----
  const float* wbp = WB + (size_t)KSB * 64 + lane * 2;
  v2f b0 = *(const v2f*)(wbp);
  v2f b1 = *(const v2f*)(wbp + 64);
#pragma unroll 1
  for (int ks = 0; ks < NKS; ks += 2) {
    v2f n0 = *(const v2f*)(wbp + (size_t)(ks + 2) * 64);   // may overrun into next
    v2f n1 = *(const v2f*)(wbp + (size_t)(ks + 3) * 64);   // chunk / zero pad: harmless
    const int koff0 = ks * 4 + 2 * half;
#pragma unroll
    for (int m = 0; m < MLO; ++m) {
      v2f a = *(const v2f*)&zt[(m * 16 + nl) * CC + koff0];
      acc[m] = __builtin_amdgcn_wmma_f32_16x16x4_f32(
          false, a, false, b0, (short)0, acc[m], false, false);
    }
#pragma unroll
    for (int m = 0; m < MLO; ++m) {
      v2f a = *(const v2f*)&zt[(m * 16 + nl) * CC + koff0 + 4];
      acc[m] = __builtin_amdgcn_wmma_f32_16x16x4_f32(
          false, a, false, b1, (short)0, acc[m], false, false);
    }
    b0 = n0;
    b1 = n1;
  }
  lds_fence_wave();   // WAR: next chunk overwrites zt (in-order LDS + compiler barrier)
}

template<int LO>
__device__ __forceinline__ void run_lo(const float* __restrict__ cg_s,
                                       const float* __restrict__ yt,
                                       const float* __restrict__ xt,
                                       float* __restrict__ zt,
                                       const float* __restrict__ WB,
                                       float* __restrict__ out,
                                       int lane, int base, int N) {
  constexpr int MLO = 2 * LO + 1;
  v8f vzero = {};
  v8f acc[MLO];
#pragma unroll
  for (int m = 0; m < MLO; ++m) acc[m] = vzero;

  if constexpr (LO == 0) {
    process_chunk<0,0,0,0,0,  0,  0>(cg_s, yt, xt, zt, WB, lane, acc);
    process_chunk<0,1,1,0,0,  1, 16>(cg_s, yt, xt, zt, WB, lane, acc);
    process_chunk<0,2,2,0,0, 10, 32>(cg_s, yt, xt, zt, WB, lane, acc);
    process_chunk<0,0,0,0,1,  0, 48>(cg_s, yt, xt, zt, WB, lane, acc);
    process_chunk<0,1,1,0,1,  1, 56>(cg_s, yt, xt, zt, WB, lane, acc);
    process_chunk<0,2,2,0,1, 10, 64>(cg_s, yt, xt, zt, WB, lane, acc);
    process_chunk<0,0,0,1,1,  0, 72>(cg_s, yt, xt, zt, WB, lane, acc);
    process_chunk<0,1,1,1,1,  1, 76>(cg_s, yt, xt, zt, WB, lane, acc);
    process_chunk<0,2,2,1,1, 10, 80>(cg_s, yt, xt, zt, WB, lane, acc);
  } else if constexpr (LO == 1) {
    process_chunk<1,0,1,0,0, 35, 84>(cg_s, yt, xt, zt, WB, lane, acc);
    process_chunk<1,1,0,0,0, 44,100>(cg_s, yt, xt, zt, WB, lane, acc);
    process_chunk<1,1,1,0,0, 53,116>(cg_s, yt, xt, zt, WB, lane, acc);
    process_chunk<1,1,2,0,0, 80,132>(cg_s, yt, xt, zt, WB, lane, acc);
    process_chunk<1,2,1,0,0,125,148>(cg_s, yt, xt, zt, WB, lane, acc);
    process_chunk<1,2,2,0,0,170,164>(cg_s, yt, xt, zt, WB, lane, acc);
    process_chunk<1,0,1,0,1, 35,180>(cg_s, yt, xt, zt, WB, lane, acc);
    process_chunk<1,1,0,0,1, 44,188>(cg_s, yt, xt, zt, WB, lane, acc);
    process_chunk<1,1,1,0,1, 53,196>(cg_s, yt, xt, zt, WB, lane, acc);
    process_chunk<1,1,2,0,1, 80,204>(cg_s, yt, xt, zt, WB, lane, acc);
    process_chunk<1,2,1,0,1,125,212>(cg_s, yt, xt, zt, WB, lane, acc);
    process_chunk<1,2,2,0,1,170,220>(cg_s, yt, xt, zt, WB, lane, acc);
    process_chunk<1,0,1,1,1, 35,228>(cg_s, yt, xt, zt, WB, lane, acc);
    process_chunk<1,1,0,1,1, 44,232>(cg_s, yt, xt, zt, WB, lane, acc);
    process_chunk<1,1,1,1,1, 53,236>(cg_s, yt, xt, zt, WB, lane, acc);
    process_chunk<1,1,2,1,1, 80,240>(cg_s, yt, xt, zt, WB, lane, acc);
    process_chunk<1,2,1,1,1,125,244>(cg_s, yt, xt, zt, WB, lane, acc);
    process_chunk<1,2,2,1,1,170,248>(cg_s, yt, xt, zt, WB, lane, acc);
  } else {
    process_chunk<2,0,2,0,0,245,252>(cg_s, yt, xt, zt, WB, lane, acc);
    process_chunk<2,1,1,0,0,270,268>(cg_s, yt, xt, zt, WB, lane, acc);
    process_chunk<2,1,2,0,0,315,284>(cg_s, yt, xt, zt, WB, lane, acc);
    process_chunk<2,2,0,0,0,390,300>(cg_s, yt, xt, zt, WB, lane, acc);
    process_chunk<2,2,1,0,0,415,316>(cg_s, yt, xt, zt, WB, lane, acc);
    process_chunk<2,2,2,0,0,490,332>(cg_s, yt, xt, zt, WB, lane, acc);
    process_chunk<2,0,2,0,1,245,348>(cg_s, yt, xt, zt, WB, lane, acc);
    process_chunk<2,1,1,0,1,270,356>(cg_s, yt, xt, zt, WB, lane, acc);
    process_chunk<2,1,2,0,1,315,364>(cg_s, yt, xt, zt, WB, lane, acc);
    process_chunk<2,2,0,0,1,390,372>(cg_s, yt, xt, zt, WB, lane, acc);
    process_chunk<2,2,1,0,1,415,380>(cg_s, yt, xt, zt, WB, lane, acc);
    process_chunk<2,2,2,0,1,490,388>(cg_s, yt, xt, zt, WB, lane, acc);
    process_chunk<2,0,2,1,1,245,396>(cg_s, yt, xt, zt, WB, lane, acc);
    process_chunk<2,1,1,1,1,270,400>(cg_s, yt, xt, zt, WB, lane, acc);
    process_chunk<2,1,2,1,1,315,404>(cg_s, yt, xt, zt, WB, lane, acc);
    process_chunk<2,2,0,1,1,390,408>(cg_s, yt, xt, zt, WB, lane, acc);
    process_chunk<2,2,1,1,1,415,412>(cg_s, yt, xt, zt, WB, lane, acc);
    process_chunk<2,2,2,1,1,490,416>(cg_s, yt, xt, zt, WB, lane, acc);
  }

  // ---- store: C/D frag vgpr r, lane l -> row M = r + 8*(l>>4), col N = l&15 ----
  const size_t ob = (LO == 0) ? 0 : (LO == 1) ? (size_t)N * 8 : (size_t)N * 32;
  const int half = lane >> 4;
  const int o = lane & 15;
  if (o < 8) {
#pragma unroll
    for (int m = 0; m < MLO; ++m)
#pragma unroll
      for (int r = 0; r < 8; ++r) {
        int gn = base + r + 8 * half;
        if (gn < N) out[ob + ((size_t)gn * 8 + o) * MLO + m] = acc[m][r];
      }
  }
}

__global__ void cgmp_main(const float* __restrict__ Y, const float* __restrict__ x0,
                          const float* __restrict__ x1, const float* __restrict__ x2,
                          const float* __restrict__ WB, const float* __restrict__ CGg,
                          float* __restrict__ out, int N) {
  extern __shared__ float smem[];
  float* cg_s = smem;                                   // 640 f32 (block-shared)
  const int tid  = threadIdx.x;
  const int wave = tid >> 5;
  const int lane = tid & 31;
  float* slab = smem + 640 + wave * SLAB_F;             // per-wave private slab
  float* yt = slab;                // [16][72]
  float* xt = slab + 16 * 72;      // [16][36]
  float* zt = xt + 16 * 36;        // [MLO][16][CC] (<= 5*16*64)

  for (int i = tid; i < 615; i += blockDim.x) cg_s[i] = CGg[i];

  const int tile = blockIdx.x * WPB + wave;
  const int base = tile * 16;
  for (int i = lane; i < 16 * 72; i += 32) {
    int node = i / 72, k = i % 72;
    int gn = base + node; if (gn >= N) gn = N - 1;
    yt[i] = Y[(size_t)gn * 72 + k];
  }
  for (int i = lane; i < 16 * 36; i += 32) {
    int node = i / 36, k = i % 36;
    int gn = base + node; if (gn >= N) gn = N - 1;
    float v;
    if (k < 4)       v = x0[(size_t)gn * 4 + k];
    else if (k < 16) v = x1[(size_t)gn * 12 + (k - 4)];
    else             v = x2[(size_t)gn * 20 + (k - 16)];
    xt[i] = v;
  }
  __syncthreads();   // cg_s is staged cooperatively across waves; tiles are wave-local

  run_lo<0>(cg_s, yt, xt, zt, WB, out, lane, base, N);
  run_lo<1>(cg_s, yt, xt, zt, WB, out, lane, base, N);
  run_lo<2>(cg_s, yt, xt, zt, WB, out, lane, base, N);
}

// ---------------- host launch ----------------
extern "C" void kernel_launch(void* const* d_in, const int* in_sizes, int n_in,
                              void* d_out, int out_size, void* d_ws, size_t ws_size,
                              hipStream_t stream) {
  const float* x0 = (const float*)d_in[0];
  const float* x1 = (const float*)d_in[1];
  const float* x2 = (const float*)d_in[2];
  const float* ev = (const float*)d_in[3];
  const float* Wxx0 = (const float*)d_in[4];
  const float* Wxx1 = (const float*)d_in[5];
  const float* Wxx2 = (const float*)d_in[6];
  const float* Wyx0 = (const float*)d_in[7];
  const float* Wyx1 = (const float*)d_in[8];
  const float* Wyx2 = (const float*)d_in[9];
  const float* Wyy0 = (const float*)d_in[10];
  const float* Wyy1 = (const float*)d_in[11];
  const float* Wyy2 = (const float*)d_in[12];
  const int* eidx = (const int*)d_in[13];
  const int N = in_sizes[0] / 4;
  const int E = in_sizes[3] / 2;

  float* wsf = (float*)d_ws;
  float* Y   = wsf;                        // N*72 f32 message accumulator
  float* CG  = wsf + (size_t)N * 72;       // 615 (pad 640) CG coefficients
  float* WB  = CG + 640;                   // (420+2)*64 f32 B-fragment weights (+pads)

  hipMemsetAsync(Y, 0, (size_t)N * 72 * sizeof(float), stream);
  cg_init<<<2, 512, 0, stream>>>(CG);
  wb_build<<<(422 * 32 + 255) / 256, 256, 0, stream>>>(
      WB, Wyy0, Wyy1, Wyy2, Wyx0, Wyx1, Wyx2, Wxx0, Wxx1, Wxx2);
  scatter_mp<<<(E + 255) / 256, 256, 0, stream>>>(x0, x1, x2, ev, eidx, Y, N, E);

  int ntiles = (N + 15) / 16;
  int blocks = (ntiles + WPB - 1) / WPB;
  size_t smem = (size_t)(640 + WPB * SLAB_F) * sizeof(float);   // ~110 KB < 320 KB/WGP
  cgmp_main<<<blocks, 128, smem, stream>>>(Y, x0, x1, x2, WB, CG, (float*)d_out, N);
}